// LearnableDemosaick_63144609186045
// MI455X (gfx1250) — compile-verified
//
#include <hip/hip_runtime.h>
#include <hip/hip_bf16.h>
#include <stdint.h>

// Problem geometry (fixed by the reference)
#define HH 1024
#define WW 1024
#define TW 64
#define TH 64
#define EPS 1e-8f

typedef __attribute__((ext_vector_type(4))) unsigned int v4u_t;
typedef __attribute__((ext_vector_type(8))) int          v8i_t;
typedef __attribute__((ext_vector_type(4))) int          v4i_t;
typedef __attribute__((ext_vector_type(4))) float        v4f_t;

static __device__ __forceinline__ int imin(int a, int b) { return a < b ? a : b; }
static __device__ __forceinline__ int imax(int a, int b) { return a > b ? a : b; }

__global__ __launch_bounds__(256)
void demosaick_tdm_kernel(const float* __restrict__ mos,
                          const float* __restrict__ gfilt,
                          const float* __restrict__ dfilt,
                          float* __restrict__ out)
{
    // LDS: raw mosaick tile with 2-px halo, and green plane with 1-px halo
    __shared__ float sM[(TH + 4) * (TW + 4)];   // 18.5 KB
    __shared__ float sG[(TH + 2) * (TW + 2)];   // 17.4 KB

    const int tid = threadIdx.x;
    const int b   = blockIdx.z;
    const int tx0 = blockIdx.x * TW;
    const int ty0 = blockIdx.y * TH;

    // Clipped load region (tile + 2-px halo, intersected with the image)
    const int lx0 = imax(tx0 - 2, 0), lx1 = imin(tx0 + TW + 2, WW);
    const int ly0 = imax(ty0 - 2, 0), ly1 = imin(ty0 + TH + 2, HH);
    const int tw  = lx1 - lx0;   // <= TW+4
    const int th  = ly1 - ly0;   // <= TH+4

    const float* src = mos + (size_t)b * (HH * WW);

    // ---- Phase 0: TDM async 2D tile load, global -> LDS (wave 0 only) ----
    if ((tid >> 5) == 0) {
        uint64_t ga  = (uint64_t)(uintptr_t)(src + (size_t)ly0 * WW + lx0);
        uint32_t lds = (uint32_t)(uintptr_t)(&sM[0]);   // flat LDS addr low 32 = LDS byte offset

        // D# group 0: count=1 valid descriptor, lds addr, 57-bit global addr, type=2 (image)
        v4u_t g0;
        g0[0] = 1u;
        g0[1] = lds;
        g0[2] = (uint32_t)ga;
        g0[3] = ((uint32_t)(ga >> 32) & 0x1FFFFFFu) | (2u << 30);

        // D# group 1: data_size=2 (4B), tensor dims == tile dims (pre-clipped: no OOB),
        // tile_dim0=tw, tile_dim1=th, dim0 stride = W elements, no pad/iterate/multicast.
        v8i_t g1;
        g1[0] = (int)(2u << 16);                                   // data_size=4B
        g1[1] = (int)(((uint32_t)tw & 0xFFFFu) << 16);             // tensor_dim0[15:0]
        g1[2] = (int)(((uint32_t)th & 0xFFFFu) << 16);             // tensor_dim0 hi=0 | tensor_dim1 lo
        g1[3] = (int)(((uint32_t)tw & 0xFFFFu) << 16);             // tensor_dim1 hi=0 | tile_dim0
        g1[4] = (int)((uint32_t)th & 0xFFFFu);                     // tile_dim1 | tile_dim2=0
        g1[5] = (int)(uint32_t)WW;                                 // tensor_dim0_stride lo32
        g1[6] = 0;                                                 // stride hi | dim1_stride lo
        g1[7] = 0;

        // D# groups 2/3: benign values for a 2D transfer (higher dims unused)
        v4i_t g2; g2[0] = 1; g2[1] = 1; g2[2] = 0; g2[3] = 0;      // tensor_dim2=1, tensor_dim3=1
        v4i_t g3; g3[0] = 0; g3[1] = (int)(1u << 16); g3[2] = 0; g3[3] = 0; // tensor_dim4=1

        // 6-arg toolchain variant: extra v8i group (zero = benign defaults), then cpol
        v8i_t gx = {0, 0, 0, 0, 0, 0, 0, 0};
        __builtin_amdgcn_tensor_load_to_lds(g0, g1, g2, g3, gx, 0);
        __builtin_amdgcn_s_wait_tensorcnt(0);
    }
    __syncthreads();   // TDM data visible to all 8 waves

    // Filter taps (uniform -> scalar loads, after the TDM load completes)
    const float gf0 = gfilt[0], gf1 = gfilt[1], gf2 = gfilt[2];
    const float df0 = dfilt[0], df1 = dfilt[1], df2 = dfilt[2];

    // LDS readers (coords are global, pre-clamped into the loaded region)
    auto ldm = [&](int cy, int cx) -> float {
        return sM[(cy - ly0) * tw + (cx - lx0)];
    };
    const int GP = TW + 2;
    auto ldg = [&](int cy, int cx) -> float {
        return sG[(cy - (ty0 - 1)) * GP + (cx - (tx0 - 1))];
    };

    // ---- Phase 1: gradient-weighted green over tile + 1-px halo ----
    for (int i = tid; i < (TH + 2) * (TW + 2); i += 256) {
        const int yy = i / GP;             // constant divisor -> mul sequence
        const int xx = i - yy * GP;
        const int gy = ty0 - 1 + yy;
        const int gx = tx0 - 1 + xx;
        // Edge clamp (matches jnp.pad mode='edge'); also keeps LDS reads in-region
        const int cy  = imin(imax(gy, 0), HH - 1);
        const int cx  = imin(imax(gx, 0), WW - 1);
        const int cyu = imax(cy - 1, 0), cyd = imin(cy + 1, HH - 1);
        const int cxl = imax(cx - 1, 0), cxr = imin(cx + 1, WW - 1);

        const float c  = ldm(cy, cx);
        const float xl = ldm(cy, cxl), xr = ldm(cy, cxr);
        const float yu = ldm(cyu, cx), yd = ldm(cyd, cx);

        const float dxv = df0 * xl + df1 * c + df2 * xr;
        const float dyv = df0 * yu + df1 * c + df2 * yd;
        const float ghv = gf0 * xl + gf1 * c + gf2 * xr;
        const float gvv = gf0 * yu + gf1 * c + gf2 * yd;

        const float adx = __builtin_fabsf(dxv);
        const float ady = __builtin_fabsf(dyv);
        const float w   = ady / (adx + ady + EPS);
        const float gi  = w * ghv + (1.0f - w) * gvv;

        const bool is_g = (((gx ^ gy) & 1) == 0);
        sG[i] = is_g ? c : gi;
    }
    __syncthreads();

    // ---- Phase 2: chroma interpolation + coalesced non-temporal float4 stores ----
    float* dstR = out + ((size_t)(b * 3 + 0) * HH) * WW;
    float* dstG = out + ((size_t)(b * 3 + 1) * HH) * WW;
    float* dstB = out + ((size_t)(b * 3 + 2) * HH) * WW;

    for (int i = tid; i < TH * (TW / 4); i += 256) {     // 4 iterations
        const int yy = i >> 4;            // TW/4 == 16 strips per row
        const int xs = (i & 15) * 4;
        const int gy = ty0 + yy;

        v4f_t r4, g4, b4;

        #pragma unroll
        for (int p = 0; p < 4; ++p) {
            const int gx = tx0 + xs + p;
            const float gc = ldg(gy, gx);
            float nr = 0.f, dr = 0.f, nb = 0.f, db = 0.f;
            #pragma unroll
            for (int dy = -1; dy <= 1; ++dy) {
                #pragma unroll
                for (int dx = -1; dx <= 1; ++dx) {
                    const int cy = gy + dy, cx = gx + dx;
                    const bool inb = (cy >= 0) & (cy < HH) & (cx >= 0) & (cx < WW);
                    const float kk = (dy == 0 ? 1.0f : 0.5f) * (dx == 0 ? 1.0f : 0.5f);
                    const bool isr = inb && (((cy & 1) == 0) & ((cx & 1) == 1));
                    const bool isb = inb && (((cy & 1) == 1) & ((cx & 1) == 0));
                    if (isr | isb) {
                        const float d = ldm(cy, cx) - ldg(cy, cx);
                        if (isr) { nr += kk * d; dr += kk; }
                        else     { nb += kk * d; db += kk; }
                    }
                }
            }
            r4[p] = gc + nr / (dr + EPS);
            g4[p] = gc;
            b4[p] = gc + nb / (db + EPS);
        }

        const size_t o = (size_t)gy * WW + (size_t)(tx0 + xs);   // 16B-aligned
        // Output is streamed exactly once -> NT hint keeps the 64 MB input
        // (which tiles re-hit via halos) resident in the 192 MB L2.
        __builtin_nontemporal_store(r4, (v4f_t*)(dstR + o));
        __builtin_nontemporal_store(g4, (v4f_t*)(dstG + o));
        __builtin_nontemporal_store(b4, (v4f_t*)(dstB + o));
    }
}

extern "C" void kernel_launch(void* const* d_in, const int* in_sizes, int n_in,
                              void* d_out, int out_size, void* d_ws, size_t ws_size,
                              hipStream_t stream) {
    const float* mos   = (const float*)d_in[0];   // [B,1,1024,1024] f32
    const float* gfilt = (const float*)d_in[1];   // [3] f32
    const float* dfilt = (const float*)d_in[2];   // [3] f32
    float*       out   = (float*)d_out;           // [B,3,1024,1024] f32

    const int B = in_sizes[0] / (HH * WW);
    dim3 grid(WW / TW, HH / TH, B);
    demosaick_tdm_kernel<<<grid, 256, 0, stream>>>(mos, gfilt, dfilt, out);
}